// LowBitMixIn_2465311228434
// MI455X (gfx1250) — compile-verified
//
#include <hip/hip_runtime.h>

// LowBitMix: out[b,o,t] = sum_{d=1..7} mixer[o,(o-d)&1023] * x[b, perm[(o-d)&1023], t]
// B=16, F_IN=F_OUT=1024, T=4096, all f32.  Memory-bound (~536 MB @ 23.3 TB/s).
//
// Banded-sparse GEMM via V_WMMA_F32_16X16X4_F32 (exact f32).
// Each wave computes four 16(o) x 16(t) tiles sharing one 16x24 A window.
// All global accesses use block-uniform SGPR bases + 32-bit lane offsets so the
// backend emits saddr/scale_offset loads (no per-row 64-bit VALU address math).

typedef float v2f __attribute__((ext_vector_type(2)));
typedef float v8f __attribute__((ext_vector_type(8)));

#define F_IN   1024
#define F_OUT  1024
#define T_DIM  4096
#define B_DIM  16

__global__ __launch_bounds__(256) void lowbit_mix_wmma_f32(
    const float* __restrict__ x,      // [B, F_IN, T]
    const float* __restrict__ mixer,  // [F_OUT, F_IN] banded, mostly zero
    const int*   __restrict__ perm,   // [F_IN]
    float*       __restrict__ out)    // [B, F_OUT, T]
{
    const int tid  = threadIdx.x;
    const int lane = tid & 31;
    const int half = lane >> 4;                 // 0: lanes 0-15, 1: lanes 16-31
    const int lr   = lane & 15;
    // wave wib handles t-span [wib*64, wib*64+64) within this block's 512 floats
    const unsigned toff = (unsigned)(((tid >> 5) << 6) + lr);

    const int tb = blockIdx.x;                  // 8 t-blocks of 512 floats
    const int ot = blockIdx.y;                  // 64 o-tiles of 16
    const int b  = blockIdx.z;                  // 16 batches
    const int o0 = ot << 4;

    // Block-uniform bases -> SGPRs; lane offsets stay 32-bit -> GVS addressing.
    const float* __restrict__ xb = x     + ((size_t)b << 22) + ((size_t)tb << 9);
    const float* __restrict__ mb = mixer + ((size_t)o0 << 10);
    float*       __restrict__ ob = out   + ((size_t)b << 22) + ((size_t)o0 << 12)
                                         + ((size_t)tb << 9);

    const unsigned arow = (unsigned)lr << 10;   // this lane's mixer row offset

    v8f acc0 = {0.f,0.f,0.f,0.f,0.f,0.f,0.f,0.f};
    v8f acc1 = acc0, acc2 = acc0, acc3 = acc0;

#pragma unroll
    for (int kk = 0; kk < 6; ++kk) {
        v2f a, b0, b1, b2, b3;
#pragma unroll
        for (int v = 0; v < 2; ++v) {
            // A[16x4]: VGPR v holds K = 2*half + v (within this 4-wide K block).
            // B[4x16]: VGPR v holds K = 2*half + v, N = lr.  Same K per lane.
            const int K = kk * 4 + half * 2 + v;
            const int j = (o0 + K - 7) & (F_IN - 1);     // cyclic column index
            a[v] = mb[arow + (unsigned)j];                // zero off-band by data
            const unsigned srow = (unsigned)perm[j];      // gathered source row
            const unsigned base = (srow << 12) + toff;    // 32-bit element offset
            b0[v] = xb[base];                             // tile 0: +0   B
            b1[v] = xb[base + 16];                        // tile 1: +64  B
            b2[v] = xb[base + 32];                        // tile 2: +128 B
            b3[v] = xb[base + 48];                        // tile 3: +192 B
        }
        // D = A*B + C, f32 exact; A reused across the 4 t-tiles.
        acc0 = __builtin_amdgcn_wmma_f32_16x16x4_f32(false, a, false, b0, (short)0, acc0, false, false);
        acc1 = __builtin_amdgcn_wmma_f32_16x16x4_f32(false, a, false, b1, (short)0, acc1, false, false);
        acc2 = __builtin_amdgcn_wmma_f32_16x16x4_f32(false, a, false, b2, (short)0, acc2, false, false);
        acc3 = __builtin_amdgcn_wmma_f32_16x16x4_f32(false, a, false, b3, (short)0, acc3, false, false);
    }

    // D layout: VGPR v -> row M = 8*half + v, col N = lr.
    const unsigned obase = ((unsigned)(half << 3) << 12) + toff;
#pragma unroll
    for (int v = 0; v < 8; ++v) {
        const unsigned r = obase + ((unsigned)v << 12);
        __builtin_nontemporal_store(acc0[v], (float*)&ob[r]);
        __builtin_nontemporal_store(acc1[v], (float*)&ob[r + 16]);
        __builtin_nontemporal_store(acc2[v], (float*)&ob[r + 32]);
        __builtin_nontemporal_store(acc3[v], (float*)&ob[r + 48]);
    }
}

extern "C" void kernel_launch(void* const* d_in, const int* in_sizes, int n_in,
                              void* d_out, int out_size, void* d_ws, size_t ws_size,
                              hipStream_t stream) {
    const float* x     = (const float*)d_in[0];
    const float* mixer = (const float*)d_in[1];
    const int*   perm  = (const int*)d_in[2];
    float*       out   = (float*)d_out;

    // grid: (t-blocks of 512, o-tiles, batches) = (8, 64, 16) -> 8192 blocks
    lowbit_mix_wmma_f32<<<dim3(T_DIM / 512, F_OUT / 16, B_DIM), dim3(256), 0, stream>>>(
        x, mixer, perm, out);
}